// Net_51994874085715
// MI455X (gfx1250) — compile-verified
//
#include <hip/hip_runtime.h>
#include <math.h>

// ---- problem constants (from reference) ----
#define N_NODES 10000
#define N_EDGES 160000
#define ETOT    (N_EDGES + N_NODES)   // self loops appended
#define FIN     50
#define C12     256
#define H12     4
#define H3      6
#define NCLS    121
#define DHID    1024                  // H12*C12
#define HD3     (H3 * NCLS)           // 726

typedef float v2f __attribute__((ext_vector_type(2)));
typedef float v4f __attribute__((ext_vector_type(4)));
typedef float v8f __attribute__((ext_vector_type(8)));

// ---------------------------------------------------------------------------
// fp32 WMMA GEMM: Out[M,Ncol] = A[M,K] @ B[K,Ncol]  (row-major)
// Block = 128 threads = 4 waves stacked on M; all waves share one 64-col tile.
// B tile (32 k x 64 cols) staged in LDS per chunk; hot loop is branch-free:
// loads use clamped addresses, only stores are guarded. K-tail handled
// separately with zero-fill (a stray k would pollute every output).
// ---------------------------------------------------------------------------
__global__ __launch_bounds__(128)
void gemm_wmma_f32(const float* __restrict__ A, const float* __restrict__ Bm,
                   float* __restrict__ Out, int M, int K, int Ncol) {
  __shared__ float Bs[32][68];                    // padded stride vs bank conflicts

  const int tid  = threadIdx.x;                   // 0..127
  const int lane = tid & 31;
  const int wave = tid >> 5;
  const int rowTile = blockIdx.y * 64 + wave * 16;
  const int colBase = blockIdx.x * 64;
  const int r    = lane & 15;
  const int hi   = lane >> 4;
  const int kOff = hi * 2;                        // lanes16-31 hold K+2,K+3

  const int rowA = min(rowTile + r, M - 1);       // clamped: loads never OOB
  const float* Ap = A + (size_t)rowA * K;

  v8f acc[4] = {v8f{}, v8f{}, v8f{}, v8f{}};

  const int Kmain = K & ~31;
  const bool vecB = (colBase + 64 <= Ncol) && ((Ncol & 3) == 0);

  for (int k0 = 0; k0 < Kmain; k0 += 32) {
    // ---- stage B[32][64] into LDS ----
    if (vecB) {
#pragma unroll
      for (int p = 0; p < 4; ++p) {
        const int idx = tid + p * 128;            // float4 slot 0..511
        const int br  = idx >> 4;                 // k row 0..31
        const int bc  = (idx & 15) * 4;           // col 0..60
        const v4f v = *(const v4f*)(Bm + (size_t)(k0 + br) * Ncol + colBase + bc);
        *(v4f*)&Bs[br][bc] = v;
      }
    } else {
#pragma unroll
      for (int p = 0; p < 16; ++p) {
        const int idx = tid + p * 128;            // float slot 0..2047
        const int br  = idx >> 6;
        const int bc  = idx & 63;
        const int col = min(colBase + bc, Ncol - 1);
        Bs[br][bc] = Bm[(size_t)(k0 + br) * Ncol + col];
      }
    }
    __syncthreads();

    // ---- 8 WMMA k-steps on the staged tile ----
#pragma unroll
    for (int kk = 0; kk < 32; kk += 4) {
      const int ka = k0 + kk + kOff;
      v2f a;
      a.x = Ap[ka];
      a.y = Ap[ka + 1];
#pragma unroll
      for (int j = 0; j < 4; ++j) {
        v2f b;
        b.x = Bs[kk + kOff    ][j * 16 + r];
        b.y = Bs[kk + kOff + 1][j * 16 + r];
        acc[j] = __builtin_amdgcn_wmma_f32_16x16x4_f32(
            false, a, false, b, (short)0, acc[j], false, false);
      }
    }
    __syncthreads();
  }

  // ---- K tail (only layer 1: K=50) -> guarded, zero-filled fragments ----
  for (int k0 = Kmain; k0 < K; k0 += 4) {
    const int ka = k0 + kOff;
    v2f a;
    a.x = (ka     < K) ? Ap[min(ka,     K - 1)] : 0.f;
    a.y = (ka + 1 < K) ? Ap[min(ka + 1, K - 1)] : 0.f;
#pragma unroll
    for (int j = 0; j < 4; ++j) {
      const int col = min(colBase + j * 16 + r, Ncol - 1);
      v2f b;
      b.x = (ka     < K) ? Bm[(size_t)min(ka,     K - 1) * Ncol + col] : 0.f;
      b.y = (ka + 1 < K) ? Bm[(size_t)min(ka + 1, K - 1) * Ncol + col] : 0.f;
      acc[j] = __builtin_amdgcn_wmma_f32_16x16x4_f32(
          false, a, false, b, (short)0, acc[j], false, false);
    }
  }

  // ---- guarded store ----
#pragma unroll
  for (int j = 0; j < 4; ++j) {
    const int col = colBase + j * 16 + r;
    if (col < Ncol) {
#pragma unroll
      for (int i = 0; i < 8; ++i) {
        const int row = rowTile + i + hi * 8;     // C/D layout: M = i (+8 hi lanes)
        if (row < M) Out[(size_t)row * Ncol + col] = acc[j][i];
      }
    }
  }
}

// ---------------------------------------------------------------------------
// Per-node attention scores: s_src[n,h] = sum_c hW[n,h,c]*a_src[h,c] (and dst)
// block = (32, heads), one wave per head, butterfly reduce (wave32).
// ---------------------------------------------------------------------------
__global__ void gat_scores(const float* __restrict__ hW,
                           const float* __restrict__ aSrc,
                           const float* __restrict__ aDst,
                           float* __restrict__ sSrc, float* __restrict__ sDst,
                           int heads, int C) {
  const int node = blockIdx.x;
  const int head = threadIdx.y;
  const int lane = threadIdx.x;
  const float* hp = hW + (size_t)node * heads * C + (size_t)head * C;
  float ss = 0.f, sd = 0.f;
  for (int c = lane; c < C; c += 32) {
    const float v = hp[c];
    ss += v * aSrc[head * C + c];
    sd += v * aDst[head * C + c];
  }
#pragma unroll
  for (int off = 16; off > 0; off >>= 1) {
    ss += __shfl_xor(ss, off, 32);
    sd += __shfl_xor(sd, off, 32);
  }
  if (lane == 0) {
    sSrc[node * heads + head] = ss;
    sDst[node * heads + head] = sd;
  }
}

__global__ void fill_f32(float* __restrict__ p, float v, long n) {
  const long i = (long)blockIdx.x * blockDim.x + threadIdx.x;
  if (i < n) p[i] = v;
}

__device__ __forceinline__ float leaky02(float x) { return x > 0.f ? x : 0.2f * x; }

__device__ __forceinline__ void atomicMaxF(float* addr, float val) {
  if (val >= 0.f) atomicMax((int*)addr, __float_as_int(val));
  else            atomicMin((unsigned int*)addr, __float_as_uint(val));
}

__global__ void edge_amax(const int* __restrict__ srcA, const int* __restrict__ dstA,
                          const float* __restrict__ sSrc, const float* __restrict__ sDst,
                          float* __restrict__ amax, int E, int Etot, int heads) {
  const long i = (long)blockIdx.x * blockDim.x + threadIdx.x;
  if (i >= (long)Etot * heads) return;
  const int h = (int)(i % heads);
  const int e = (int)(i / heads);
  const int s = (e < E) ? srcA[e] : (e - E);
  const int d = (e < E) ? dstA[e] : (e - E);
  const float alpha = leaky02(sSrc[s * heads + h] + sDst[d * heads + h]);
  atomicMaxF(&amax[d * heads + h], alpha);
}

__global__ void edge_exp(const int* __restrict__ srcA, const int* __restrict__ dstA,
                         const float* __restrict__ sSrc, const float* __restrict__ sDst,
                         const float* __restrict__ amax, float* __restrict__ denom,
                         float* __restrict__ exb, int E, int Etot, int heads) {
  const long i = (long)blockIdx.x * blockDim.x + threadIdx.x;
  if (i >= (long)Etot * heads) return;
  const int h = (int)(i % heads);
  const int e = (int)(i / heads);
  const int s = (e < E) ? srcA[e] : (e - E);
  const int d = (e < E) ? dstA[e] : (e - E);
  const float alpha = leaky02(sSrc[s * heads + h] + sDst[d * heads + h]);
  const float ex = expf(alpha - amax[d * heads + h]);
  exb[(size_t)e * heads + h] = ex;
  atomicAdd(&denom[d * heads + h], ex);
}

// Heavy kernel: agg[dst] += hW[src] * w  (L2-resident: 40MB << 192MB L2)
__global__ __launch_bounds__(256)
void edge_aggregate(const int* __restrict__ srcA, const int* __restrict__ dstA,
                    const float* __restrict__ hW, const float* __restrict__ exb,
                    const float* __restrict__ denom, float* __restrict__ agg,
                    int E, int heads, int C, int HD) {
  const int e = blockIdx.x;
  const int s = (e < E) ? srcA[e] : (e - E);
  const int d = (e < E) ? dstA[e] : (e - E);
  __shared__ float wsh[8];
  if ((int)threadIdx.x < heads)
    wsh[threadIdx.x] = exb[(size_t)e * heads + threadIdx.x] /
                       (denom[d * heads + threadIdx.x] + 1e-16f);
  __syncthreads();
  const float* hp = hW + (size_t)s * HD;
  float* ap = agg + (size_t)d * HD;
  for (int h = 0; h < heads; ++h) {
    const float w = wsh[h];
    const float* hph = hp + h * C;
    float* aph = ap + h * C;
    for (int c = threadIdx.x; c < C; c += blockDim.x)
      atomicAdd(&aph[c], hph[c] * w);
  }
}

// h = elu(agg + b + lin + lb)    (layers 1 & 2)
__global__ void combine_elu(const float* __restrict__ agg, const float* __restrict__ b,
                            const float* __restrict__ lin, const float* __restrict__ lb,
                            float* __restrict__ out, long total, int Dd) {
  const long i = (long)blockIdx.x * blockDim.x + threadIdx.x;
  if (i >= total) return;
  const int c = (int)(i % Dd);
  const float v = agg[i] + b[c] + lin[i] + lb[c];
  out[i] = v > 0.f ? v : expm1f(v);
}

// out = mean_h(agg3) + b3 + lin3 + lb3   (layer 3, concat=False)
__global__ void combine_final(const float* __restrict__ agg3, const float* __restrict__ b3,
                              const float* __restrict__ lin3, const float* __restrict__ lb3,
                              float* __restrict__ out) {
  const long i = (long)blockIdx.x * blockDim.x + threadIdx.x;
  if (i >= (long)N_NODES * NCLS) return;
  const int n = (int)(i / NCLS);
  const int c = (int)(i % NCLS);
  float s = 0.f;
#pragma unroll
  for (int h = 0; h < H3; ++h) s += agg3[(size_t)n * HD3 + h * NCLS + c];
  out[i] = s * (1.f / H3) + b3[c] + lin3[i] + lb3[c];
}

// ---------------------------------------------------------------------------
static void gat_layer(const float* feat, int K, const float* Wm,
                      const float* aS, const float* aD,
                      int heads, int C, int HD,
                      const int* srcA, const int* dstA,
                      float* hW, float* sS, float* sD, float* amax,
                      float* denom, float* exb, float* agg, hipStream_t stream) {
  dim3 gg((HD + 63) / 64, (N_NODES + 63) / 64);
  gemm_wmma_f32<<<gg, 128, 0, stream>>>(feat, Wm, hW, N_NODES, K, HD);

  gat_scores<<<N_NODES, dim3(32, heads), 0, stream>>>(hW, aS, aD, sS, sD, heads, C);

  const long nh = (long)N_NODES * heads;
  fill_f32<<<(int)((nh + 255) / 256), 256, 0, stream>>>(amax, -INFINITY, nh);
  fill_f32<<<(int)((nh + 255) / 256), 256, 0, stream>>>(denom, 0.f, nh);
  const long na = (long)N_NODES * HD;
  fill_f32<<<(int)((na + 255) / 256), 256, 0, stream>>>(agg, 0.f, na);

  const long te = (long)ETOT * heads;
  edge_amax<<<(int)((te + 255) / 256), 256, 0, stream>>>(srcA, dstA, sS, sD, amax,
                                                         N_EDGES, ETOT, heads);
  edge_exp<<<(int)((te + 255) / 256), 256, 0, stream>>>(srcA, dstA, sS, sD, amax, denom,
                                                        exb, N_EDGES, ETOT, heads);
  edge_aggregate<<<ETOT, 256, 0, stream>>>(srcA, dstA, hW, exb, denom, agg,
                                           N_EDGES, heads, C, HD);
}

extern "C" void kernel_launch(void* const* d_in, const int* in_sizes, int n_in,
                              void* d_out, int out_size, void* d_ws, size_t ws_size,
                              hipStream_t stream) {
  const float* x      = (const float*)d_in[0];
  const int*   ei     = (const int*)d_in[1];
  const int*   srcA   = ei;
  const int*   dstA   = ei + N_EDGES;
  const float* W1     = (const float*)d_in[2];
  const float* a_src1 = (const float*)d_in[3];
  const float* a_dst1 = (const float*)d_in[4];
  const float* b1     = (const float*)d_in[5];
  const float* lW1    = (const float*)d_in[6];
  const float* lb1    = (const float*)d_in[7];
  const float* W2     = (const float*)d_in[8];
  const float* a_src2 = (const float*)d_in[9];
  const float* a_dst2 = (const float*)d_in[10];
  const float* b2     = (const float*)d_in[11];
  const float* lW2    = (const float*)d_in[12];
  const float* lb2    = (const float*)d_in[13];
  const float* W3     = (const float*)d_in[14];
  const float* a_src3 = (const float*)d_in[15];
  const float* a_dst3 = (const float*)d_in[16];
  const float* b3     = (const float*)d_in[17];
  const float* lW3    = (const float*)d_in[18];
  const float* lb3    = (const float*)d_in[19];

  float* ws = (float*)d_ws;
  const size_t NB = (size_t)N_NODES * DHID;
  float* B0    = ws;            // hW projection
  float* B1    = ws + NB;       // linear branch
  float* B2    = ws + 2 * NB;   // agg (segment-sum output)
  float* B3    = ws + 3 * NB;   // current node features h
  float* sS    = ws + 4 * NB;
  float* sD    = sS + (size_t)N_NODES * 8;
  float* amax  = sD + (size_t)N_NODES * 8;
  float* denom = amax + (size_t)N_NODES * 8;
  float* exb   = denom + (size_t)N_NODES * 8;   // ETOT*8 floats

  // -------- layer 1 --------
  gat_layer(x, FIN, W1, a_src1, a_dst1, H12, C12, DHID,
            srcA, dstA, B0, sS, sD, amax, denom, exb, B2, stream);
  {
    dim3 gg((DHID + 63) / 64, (N_NODES + 63) / 64);
    gemm_wmma_f32<<<gg, 128, 0, stream>>>(x, lW1, B1, N_NODES, FIN, DHID);
  }
  {
    const long t = (long)N_NODES * DHID;
    combine_elu<<<(int)((t + 255) / 256), 256, 0, stream>>>(B2, b1, B1, lb1, B3, t, DHID);
  }

  // -------- layer 2 --------
  gat_layer(B3, DHID, W2, a_src2, a_dst2, H12, C12, DHID,
            srcA, dstA, B0, sS, sD, amax, denom, exb, B2, stream);
  {
    dim3 gg((DHID + 63) / 64, (N_NODES + 63) / 64);
    gemm_wmma_f32<<<gg, 128, 0, stream>>>(B3, lW2, B1, N_NODES, DHID, DHID);
  }
  {
    const long t = (long)N_NODES * DHID;
    combine_elu<<<(int)((t + 255) / 256), 256, 0, stream>>>(B2, b2, B1, lb2, B3, t, DHID);
  }

  // -------- layer 3 --------
  gat_layer(B3, DHID, W3, a_src3, a_dst3, H3, NCLS, HD3,
            srcA, dstA, B0, sS, sD, amax, denom, exb, B2, stream);
  {
    dim3 gg((NCLS + 63) / 64, (N_NODES + 63) / 64);
    gemm_wmma_f32<<<gg, 128, 0, stream>>>(B3, lW3, B1, N_NODES, DHID, NCLS);
  }
  {
    const long t = (long)N_NODES * NCLS;
    combine_final<<<(int)((t + 255) / 256), 256, 0, stream>>>(B2, b3, B1, lb3, (float*)d_out);
  }
}